// DGCNNEncoder_18098992185534
// MI455X (gfx1250) — compile-verified
//
#include <hip/hip_runtime.h>

typedef __attribute__((ext_vector_type(2))) float v2f;
typedef __attribute__((ext_vector_type(8))) float v8f;

#define BATCH 8
#define NPTS  1024
#define KNN_K 20

__device__ __forceinline__ v8f wmma_f32_16x16x4(v2f a, v2f b, v8f c) {
  // D = A(16x4) * B(4x16) + C, fp32, wave32.
  return __builtin_amdgcn_wmma_f32_16x16x4_f32(false, a, false, b, (short)0, c, false, false);
}

__device__ __forceinline__ unsigned sortable_encode(float f) {
  unsigned u = __float_as_uint(f);
  return (u & 0x80000000u) ? ~u : (u | 0x80000000u);
}
__device__ __forceinline__ float sortable_decode(unsigned u) {
  unsigned bits = (u & 0x80000000u) ? (u & 0x7fffffffu) : ~u;
  return __uint_as_float(bits);
}

// ---------------------------------------------------------------- utilities
__global__ void fill_u32(unsigned* __restrict__ p, unsigned val, int n) {
  int t = blockIdx.x * blockDim.x + threadIdx.x;
  if (t < n) p[t] = val;
}

// Stage x (B,N,3)->(B,N,4 padded) and w1 (64,6)->(64,8 padded edge layout)
__global__ void pack_kernel(const float* __restrict__ x, const float* __restrict__ w1,
                            float* __restrict__ F0, float* __restrict__ W1P) {
  int t = blockIdx.x * blockDim.x + threadIdx.x;
  if (t < BATCH * NPTS) {
    F0[t * 4 + 0] = x[t * 3 + 0];
    F0[t * 4 + 1] = x[t * 3 + 1];
    F0[t * 4 + 2] = x[t * 3 + 2];
    F0[t * 4 + 3] = 0.f;
  }
  if (t < 64) {
#pragma unroll
    for (int c = 0; c < 8; ++c) {
      float v = 0.f;
      if (c < 3) v = w1[t * 6 + c];            // neigh-ctr part
      else if (c >= 4 && c < 7) v = w1[t * 6 + c - 1]; // ctr part
      W1P[t * 8 + c] = v;
    }
  }
}

template <int Cp>
__global__ void sqnorm_kernel(const float* __restrict__ f, float* __restrict__ sq) {
  int t = blockIdx.x * blockDim.x + threadIdx.x;
  if (t < BATCH * NPTS) {
    float s = 0.f;
#pragma unroll 8
    for (int c = 0; c < Cp; ++c) { float v = f[(size_t)t * Cp + c]; s += v * v; }
    sq[t] = s;
  }
}

// ------------------------------------------------- pairwise (neg sq dist) GEMM
// One wave computes a 16x16 tile of P[b][n][m] = 2*<f_n,f_m> - |f_n|^2 - |f_m|^2
template <int Cp>
__global__ __launch_bounds__(128) void pair_kernel(const float* __restrict__ f,
                                                   const float* __restrict__ sq,
                                                   float* __restrict__ P) {
  const int b = blockIdx.y;
  const int wave = threadIdx.x >> 5;
  const int lane = threadIdx.x & 31;
  const int half = lane >> 4, l16 = lane & 15;
  const int tile = blockIdx.x * 4 + wave;        // 0..4095
  const int n0 = (tile >> 6) << 4;
  const int m0 = (tile & 63) << 4;
  const float* fb = f + (size_t)b * NPTS * Cp;
  const float* arow = fb + (size_t)(n0 + l16) * Cp + 2 * half; // A: rows n, k-pairs
  const float* brow = fb + (size_t)(m0 + l16) * Cp + 2 * half; // B: cols m, k-pairs
  v8f acc = {};
#pragma unroll 8
  for (int kb = 0; kb < Cp; kb += 4) {
    v2f a  = *(const v2f*)(arow + kb);
    v2f bb = *(const v2f*)(brow + kb);
    acc = wmma_f32_16x16x4(a, bb, acc);
  }
  const float* sqb = sq + b * NPTS;
  const float sqm = sqb[m0 + l16];
  float* Pb = P + ((size_t)b * NPTS + n0) * NPTS + m0;
#pragma unroll
  for (int v = 0; v < 8; ++v) {
    int row = v + half * 8;
    Pb[(size_t)row * NPTS + l16] = 2.f * acc[v] - sqb[n0 + row] - sqm;
  }
}

// ------------------------------------------------------------------- top-k
__global__ __launch_bounds__(256) void topk_kernel(const float* __restrict__ P,
                                                   int* __restrict__ knn) {
  __shared__ float vals[NPTS];
  __shared__ float rv[256];
  __shared__ int   ri[256];
  const int row = blockIdx.x;               // b*NPTS + n
  const int t = threadIdx.x;
  const float* pr = P + (size_t)row * NPTS;
  for (int i = t; i < NPTS; i += 256) vals[i] = pr[i];
  __syncthreads();
  int* outk = knn + (size_t)row * KNN_K;
  for (int s = 0; s < KNN_K; ++s) {
    float bv = -__builtin_inff(); int bi = NPTS;
    for (int i = t; i < NPTS; i += 256) {
      float v = vals[i];
      if (v > bv || (v == bv && i < bi)) { bv = v; bi = i; }
    }
    rv[t] = bv; ri[t] = bi;
    __syncthreads();
    for (int off = 128; off > 0; off >>= 1) {
      if (t < off) {
        float v2 = rv[t + off]; int i2 = ri[t + off];
        if (v2 > rv[t] || (v2 == rv[t] && i2 < ri[t])) { rv[t] = v2; ri[t] = i2; }
      }
      __syncthreads();
    }
    if (t == 0) { outk[s] = ri[0]; vals[ri[0]] = -__builtin_inff(); }
    __syncthreads();
  }
}

// ---------------------------------------------- fused gather + edge-conv GEMM
// STATS=true : accumulate per-channel sum / sumsq of y over (n-tile, k)
// STATS=false: apply BN scale/shift + leaky-relu, max over k, write (B,N,O)
template <int Cp, int O, bool STATS>
__global__ __launch_bounds__(128) void edge_conv_kernel(
    const float* __restrict__ f, const int* __restrict__ knn,
    const float* __restrict__ W, float* __restrict__ out,
    float* __restrict__ d_sum, float* __restrict__ d_sq,
    const float* __restrict__ scale, const float* __restrict__ shift) {
  constexpr int KT = 2 * Cp;
  __shared__ v2f E[KT / 2][16];              // edge tile, k-pairs x 16 points
  const int b  = blockIdx.x >> 6;
  const int n0 = (blockIdx.x & 63) << 4;
  const int wave = threadIdx.x >> 5;
  const int lane = threadIdx.x & 31;
  const int half = lane >> 4, l16 = lane & 15;
  const int mt = blockIdx.y * 4 + wave;      // output-channel tile
  const int o0 = mt * 16;
  const float* fb = f + (size_t)b * NPTS * Cp;
  const int* kr = knn + ((size_t)b * NPTS + n0) * KNN_K;

  float sc[8], sh[8];
  if (!STATS) {
#pragma unroll
    for (int v = 0; v < 8; ++v) { int o = o0 + v + 8 * half; sc[v] = scale[o]; sh[v] = shift[o]; }
  }
  v8f accA, accB;
#pragma unroll
  for (int v = 0; v < 8; ++v) { accA[v] = STATS ? 0.f : -__builtin_inff(); accB[v] = 0.f; }

  const float* wrow = W + (size_t)(o0 + l16) * KT + 2 * half;

  for (int kk = 0; kk < KNN_K; ++kk) {
    // cooperative build of edge features [neigh-ctr | ctr] into LDS
    for (int e = threadIdx.x; e < (KT / 2) * 16; e += 128) {
      int col = e & 15;
      int kp  = e >> 4;
      int c0  = kp * 2;
      int n   = n0 + col;
      int j   = kr[col * KNN_K + kk];
      float v0, v1;
      if (c0 < Cp) {
        v0 = fb[(size_t)j * Cp + c0]     - fb[(size_t)n * Cp + c0];
        v1 = fb[(size_t)j * Cp + c0 + 1] - fb[(size_t)n * Cp + c0 + 1];
      } else {
        v0 = fb[(size_t)n * Cp + (c0 - Cp)];
        v1 = fb[(size_t)n * Cp + (c0 - Cp) + 1];
      }
      v2f t; t.x = v0; t.y = v1;
      E[kp][col] = t;
    }
    __syncthreads();

    v8f c = {};
#pragma unroll 8
    for (int k4 = 0; k4 < KT; k4 += 4) {
      v2f a  = *(const v2f*)(wrow + k4);
      v2f bb = E[(k4 >> 1) + half][l16];
      c = wmma_f32_16x16x4(a, bb, c);
    }
    if (STATS) {
#pragma unroll
      for (int v = 0; v < 8; ++v) { accA[v] += c[v]; accB[v] += c[v] * c[v]; }
    } else {
#pragma unroll
      for (int v = 0; v < 8; ++v) {
        float yn = c[v] * sc[v] + sh[v];
        yn = yn >= 0.f ? yn : 0.2f * yn;
        accA[v] = fmaxf(accA[v], yn);
      }
    }
    __syncthreads();
  }

  if (STATS) {
#pragma unroll
    for (int v = 0; v < 8; ++v) {
      float s = accA[v], q = accB[v];
#pragma unroll
      for (int off = 8; off > 0; off >>= 1) {
        s += __shfl_xor(s, off, 16);
        q += __shfl_xor(q, off, 16);
      }
      if (l16 == 0) {
        int o = o0 + v + 8 * half;
        atomicAdd(&d_sum[o], s);
        atomicAdd(&d_sq[o], q);
      }
    }
  } else {
    float* ob = out + ((size_t)b * NPTS + n0 + l16) * O + o0 + 8 * half;
#pragma unroll
    for (int v = 0; v < 8; ++v) ob[v] = accA[v];
  }
}

__global__ void bn_finalize(const float* __restrict__ sum, const float* __restrict__ sq,
                            const float* __restrict__ g, const float* __restrict__ bb,
                            float* __restrict__ scale, float* __restrict__ shift,
                            int O, float invcnt) {
  int o = blockIdx.x * blockDim.x + threadIdx.x;
  if (o < O) {
    float m = sum[o] * invcnt;
    float var = sq[o] * invcnt - m * m;
    float s = g[o] * rsqrtf(var + 1e-5f);
    scale[o] = s;
    shift[o] = bb[o] - m * s;
  }
}

// ------------------------------------- final 512-ch GEMM over concat features
template <bool STATS>
__global__ __launch_bounds__(128) void final_conv_kernel(
    const float* __restrict__ f1, const float* __restrict__ f2,
    const float* __restrict__ f3, const float* __restrict__ f4,
    const float* __restrict__ W5,
    float* __restrict__ d_sum, float* __restrict__ d_sq,
    const float* __restrict__ scale, const float* __restrict__ shift,
    unsigned* __restrict__ gmax) {
  __shared__ v2f E[256][16];                 // 512 channels as 256 pairs x 16 pts
  const int b  = blockIdx.x >> 6;
  const int n0 = (blockIdx.x & 63) << 4;
  const int wave = threadIdx.x >> 5, lane = threadIdx.x & 31;
  const int half = lane >> 4, l16 = lane & 15;
  for (int e = threadIdx.x; e < 256 * 16; e += 128) {
    int col = e & 15, kp = e >> 4, c0 = kp * 2;
    size_t n = (size_t)b * NPTS + n0 + col;
    const float* src;
    if      (c0 < 64)  src = f1 + n * 64  + c0;
    else if (c0 < 128) src = f2 + n * 64  + (c0 - 64);
    else if (c0 < 256) src = f3 + n * 128 + (c0 - 128);
    else               src = f4 + n * 256 + (c0 - 256);
    v2f t; t.x = src[0]; t.y = src[1];
    E[kp][col] = t;
  }
  __syncthreads();
  const int mt = blockIdx.y * 4 + wave;
  const int o0 = mt * 16;
  const float* wrow = W5 + (size_t)(o0 + l16) * 512 + 2 * half;
  v8f c = {};
#pragma unroll 8
  for (int k4 = 0; k4 < 512; k4 += 4) {
    v2f a  = *(const v2f*)(wrow + k4);
    v2f bb = E[(k4 >> 1) + half][l16];
    c = wmma_f32_16x16x4(a, bb, c);
  }
  if (STATS) {
#pragma unroll
    for (int v = 0; v < 8; ++v) {
      float s = c[v], q = c[v] * c[v];
#pragma unroll
      for (int off = 8; off > 0; off >>= 1) {
        s += __shfl_xor(s, off, 16);
        q += __shfl_xor(q, off, 16);
      }
      if (l16 == 0) {
        int o = o0 + v + 8 * half;
        atomicAdd(&d_sum[o], s);
        atomicAdd(&d_sq[o], q);
      }
    }
  } else {
#pragma unroll
    for (int v = 0; v < 8; ++v) {
      int o = o0 + v + 8 * half;
      float yn = c[v] * scale[o] + shift[o];
      yn = yn >= 0.f ? yn : 0.2f * yn;
#pragma unroll
      for (int off = 8; off > 0; off >>= 1) yn = fmaxf(yn, __shfl_xor(yn, off, 16));
      if (l16 == 0) atomicMax(&gmax[b * 512 + o], sortable_encode(yn));
    }
  }
}

// out[b][of] = sum_o gmax[b][o] * wemb[of][o]
__global__ void emb_kernel(const unsigned* __restrict__ gmax,
                           const float* __restrict__ wemb, float* __restrict__ out) {
  int t = blockIdx.x * blockDim.x + threadIdx.x;
  if (t < BATCH * 256) {
    int b = t >> 8, of = t & 255;
    const unsigned* gb = gmax + b * 512;
    const float* wr = wemb + (size_t)of * 512;
    float s = 0.f;
    for (int o = 0; o < 512; ++o) s += sortable_decode(gb[o]) * wr[o];
    out[t] = s;
  }
}

// ---------------------------------------------------------------------------
extern "C" void kernel_launch(void* const* d_in, const int* in_sizes, int n_in,
                              void* d_out, int out_size, void* d_ws, size_t ws_size,
                              hipStream_t stream) {
  (void)in_sizes; (void)n_in; (void)out_size; (void)ws_size;
  const float* x    = (const float*)d_in[0];
  const float* w1   = (const float*)d_in[1];
  const float* g1   = (const float*)d_in[2];
  const float* b1   = (const float*)d_in[3];
  const float* w2   = (const float*)d_in[4];
  const float* g2   = (const float*)d_in[5];
  const float* b2   = (const float*)d_in[6];
  const float* w3   = (const float*)d_in[7];
  const float* g3   = (const float*)d_in[8];
  const float* b3   = (const float*)d_in[9];
  const float* w4   = (const float*)d_in[10];
  const float* g4   = (const float*)d_in[11];
  const float* b4   = (const float*)d_in[12];
  const float* w5   = (const float*)d_in[13];
  const float* g5   = (const float*)d_in[14];
  const float* b5   = (const float*)d_in[15];
  const float* wemb = (const float*)d_in[16];
  float* out = (float*)d_out;

  float* p = (float*)d_ws;
  float* F0  = p; p += (size_t)BATCH * NPTS * 4;
  float* F1  = p; p += (size_t)BATCH * NPTS * 64;
  float* F2  = p; p += (size_t)BATCH * NPTS * 64;
  float* F3  = p; p += (size_t)BATCH * NPTS * 128;
  float* F4  = p; p += (size_t)BATCH * NPTS * 256;
  float* P   = p; p += (size_t)BATCH * NPTS * NPTS;
  float* SQ  = p; p += (size_t)BATCH * NPTS;
  float* W1P = p; p += 64 * 8;
  float* SUM = p; p += 512;   // SUM and SQS contiguous: zeroed together
  float* SQS = p; p += 512;
  float* SCL = p; p += 512;
  float* SHF = p; p += 512;
  unsigned* GMAX = (unsigned*)p; p += BATCH * 512;
  int* KNNI = (int*)p;

  const float invE = 1.0f / (float)(BATCH * NPTS * KNN_K); // edge BN count
  const float invF = 1.0f / (float)(BATCH * NPTS);         // final BN count

  pack_kernel<<<(BATCH * NPTS + 255) / 256, 256, 0, stream>>>(x, w1, F0, W1P);

  // ---- block 1 : Cin 3 (pad 4) -> 64
  sqnorm_kernel<4><<<(BATCH * NPTS + 255) / 256, 256, 0, stream>>>(F0, SQ);
  pair_kernel<4><<<dim3(1024, BATCH), 128, 0, stream>>>(F0, SQ, P);
  topk_kernel<<<BATCH * NPTS, 256, 0, stream>>>(P, KNNI);
  fill_u32<<<4, 256, 0, stream>>>((unsigned*)SUM, 0u, 1024);
  edge_conv_kernel<4, 64, true><<<dim3(512, 1), 128, 0, stream>>>(F0, KNNI, W1P, nullptr, SUM, SQS, nullptr, nullptr);
  bn_finalize<<<1, 64, 0, stream>>>(SUM, SQS, g1, b1, SCL, SHF, 64, invE);
  edge_conv_kernel<4, 64, false><<<dim3(512, 1), 128, 0, stream>>>(F0, KNNI, W1P, F1, nullptr, nullptr, SCL, SHF);

  // ---- block 2 : 64 -> 64
  sqnorm_kernel<64><<<(BATCH * NPTS + 255) / 256, 256, 0, stream>>>(F1, SQ);
  pair_kernel<64><<<dim3(1024, BATCH), 128, 0, stream>>>(F1, SQ, P);
  topk_kernel<<<BATCH * NPTS, 256, 0, stream>>>(P, KNNI);
  fill_u32<<<4, 256, 0, stream>>>((unsigned*)SUM, 0u, 1024);
  edge_conv_kernel<64, 64, true><<<dim3(512, 1), 128, 0, stream>>>(F1, KNNI, w2, nullptr, SUM, SQS, nullptr, nullptr);
  bn_finalize<<<1, 64, 0, stream>>>(SUM, SQS, g2, b2, SCL, SHF, 64, invE);
  edge_conv_kernel<64, 64, false><<<dim3(512, 1), 128, 0, stream>>>(F1, KNNI, w2, F2, nullptr, nullptr, SCL, SHF);

  // ---- block 3 : 64 -> 128
  sqnorm_kernel<64><<<(BATCH * NPTS + 255) / 256, 256, 0, stream>>>(F2, SQ);
  pair_kernel<64><<<dim3(1024, BATCH), 128, 0, stream>>>(F2, SQ, P);
  topk_kernel<<<BATCH * NPTS, 256, 0, stream>>>(P, KNNI);
  fill_u32<<<4, 256, 0, stream>>>((unsigned*)SUM, 0u, 1024);
  edge_conv_kernel<64, 128, true><<<dim3(512, 2), 128, 0, stream>>>(F2, KNNI, w3, nullptr, SUM, SQS, nullptr, nullptr);
  bn_finalize<<<2, 64, 0, stream>>>(SUM, SQS, g3, b3, SCL, SHF, 128, invE);
  edge_conv_kernel<64, 128, false><<<dim3(512, 2), 128, 0, stream>>>(F2, KNNI, w3, F3, nullptr, nullptr, SCL, SHF);

  // ---- block 4 : 128 -> 256
  sqnorm_kernel<128><<<(BATCH * NPTS + 255) / 256, 256, 0, stream>>>(F3, SQ);
  pair_kernel<128><<<dim3(1024, BATCH), 128, 0, stream>>>(F3, SQ, P);
  topk_kernel<<<BATCH * NPTS, 256, 0, stream>>>(P, KNNI);
  fill_u32<<<4, 256, 0, stream>>>((unsigned*)SUM, 0u, 1024);
  edge_conv_kernel<128, 256, true><<<dim3(512, 4), 128, 0, stream>>>(F3, KNNI, w4, nullptr, SUM, SQS, nullptr, nullptr);
  bn_finalize<<<4, 64, 0, stream>>>(SUM, SQS, g4, b4, SCL, SHF, 256, invE);
  edge_conv_kernel<128, 256, false><<<dim3(512, 4), 128, 0, stream>>>(F3, KNNI, w4, F4, nullptr, nullptr, SCL, SHF);

  // ---- final : concat(512) GEMM + BN + lrelu + global max over N + embed
  fill_u32<<<4, 256, 0, stream>>>((unsigned*)SUM, 0u, 1024);
  final_conv_kernel<true><<<dim3(512, 8), 128, 0, stream>>>(F1, F2, F3, F4, w5, SUM, SQS, nullptr, nullptr, nullptr);
  bn_finalize<<<8, 64, 0, stream>>>(SUM, SQS, g5, b5, SCL, SHF, 512, invF);
  fill_u32<<<(BATCH * 512 + 255) / 256, 256, 0, stream>>>(GMAX, 0u, BATCH * 512);
  final_conv_kernel<false><<<dim3(512, 8), 128, 0, stream>>>(F1, F2, F3, F4, w5, nullptr, nullptr, SCL, SHF, GMAX);
  emb_kernel<<<(BATCH * 256 + 255) / 256, 256, 0, stream>>>(GMAX, wemb, out);
}